// TwoStageModel_26371099197503
// MI455X (gfx1250) — compile-verified
//
#include <hip/hip_runtime.h>
#include <cstdint>
#include <cstddef>

typedef float v2f __attribute__((ext_vector_type(2)));
typedef float v8f __attribute__((ext_vector_type(8)));

#define NSEG    32          // column segments (grid.y)
#define LISTN   128         // per-row running "best so far" list size (>= TOPK)
#define BUFCAP  2048        // worst case per round: 8 waves * 32 lanes * 8 vals
#define NEG_INF (-3.402823466e38f)

// ---------------------------------------------------------------------------
// Kernel 1: fused f32-WMMA GEMM + streaming per-row 128-best selection.
// Grid: (B/16 row tiles, NSEG segments). Block: 256 threads (8 wave32s).
// Each wave computes one 16x16 score tile per round via 16x V_WMMA_F32_16X16X4_F32,
// split across two independent accumulator chains to hide XDL pipeline latency.
// ---------------------------------------------------------------------------
__global__ __launch_bounds__(256, 1)
void topk_gemm_partial(const float* __restrict__ user,
                       const float* __restrict__ corpus,
                       float* __restrict__ cand_val,
                       unsigned* __restrict__ cand_idx,
                       int N, int segcols)
{
    const int tid  = threadIdx.x;
    const int wave = tid >> 5;
    const int lane = tid & 31;
    const int rowTile  = blockIdx.x;        // 0..15 (16 user rows each)
    const int seg      = blockIdx.y;        // 0..NSEG-1
    const int segStart = seg * segcols;
    const int segEnd   = min(N, segStart + segcols);

    __shared__ float    Asm[16][65];            // user tile, padded vs bank conflicts
    __shared__ float    lv[16][LISTN];          // per-row best-128 values
    __shared__ unsigned li[16][LISTN];          // per-row best-128 indices
    __shared__ float    lmin[16];               // cached min of each list (threshold)
    __shared__ int      lminpos[16];
    __shared__ float    bufv[BUFCAP];           // candidate buffer (per round)
    __shared__ unsigned bufm[BUFCAP];           // packed (row<<26 | col)
    __shared__ int      cnt;

    // stage A tile + init lists
    for (int i = tid; i < 16 * 64; i += 256)
        Asm[i >> 6][i & 63] = user[(size_t)(rowTile * 16 + (i >> 6)) * 64 + (i & 63)];
    for (int i = tid; i < 16 * LISTN; i += 256) {
        lv[i >> 7][i & 127] = NEG_INF;
        li[i >> 7][i & 127] = 0u;
    }
    if (tid < 16) { lmin[tid] = NEG_INF; lminpos[tid] = 0; }
    if (tid == 0) cnt = 0;
    __syncthreads();

    // Build A fragments (16x4 f32 per chunk): lanes 0-15 -> K={4c,4c+1},
    // lanes 16-31 -> K={4c+2,4c+3}, M = lane%16.
    const int mrow = lane & 15;
    const int half = lane >> 4;
    v2f afrag[16];
#pragma unroll
    for (int c = 0; c < 16; ++c) {
        const int kb = 4 * c + 2 * half;
        v2f a; a[0] = Asm[mrow][kb]; a[1] = Asm[mrow][kb + 1];
        afrag[c] = a;
    }

    const int ntiles = (segEnd - segStart + 15) >> 4;  // 16-col tiles in segment
    const int rounds = (ntiles + 7) >> 3;              // 8 waves per round

    for (int r = 0; r < rounds; ++r) {
        const int t = r * 8 + wave;
        if (t < ntiles) {
            const int col0 = segStart + t * 16;
            const int col  = col0 + (lane & 15);       // this lane's corpus row
            const int nc   = min(col, N - 1);
            const float* cp = corpus + (size_t)nc * 64 + 2 * half;

            // prefetch next round's tile for this wave
            const int tn = t + 8;
            if (tn < ntiles)
                __builtin_prefetch(corpus + (size_t)(segStart + tn * 16 + (lane & 15)) * 64, 0, 0);

            // B fragments (4x16 f32 per chunk), mirrored layout of A
            v2f bfrag[16];
#pragma unroll
            for (int c = 0; c < 16; ++c)
                bfrag[c] = *(const v2f*)(cp + 4 * c);

            // Two independent accumulation chains -> interleaved WMMA issue
            v8f acc0 = {}, acc1 = {};
#pragma unroll
            for (int c = 0; c < 8; ++c) {
                acc0 = __builtin_amdgcn_wmma_f32_16x16x4_f32(
                    false, afrag[c],     false, bfrag[c],     (short)0, acc0, false, false);
                acc1 = __builtin_amdgcn_wmma_f32_16x16x4_f32(
                    false, afrag[c + 8], false, bfrag[c + 8], (short)0, acc1, false, false);
            }
            const v8f acc = acc0 + acc1;

            // C layout: VGPR k, lanes 0-15 -> M=k, lanes 16-31 -> M=k+8; N = lane%16
            const bool colok = (col < segEnd);
            const int  mbase = half * 8;
#pragma unroll
            for (int k = 0; k < 8; ++k) {
                const int   m = mbase + k;
                const float v = acc[k];
                if (colok && v > lmin[m]) {
                    const int p = atomicAdd(&cnt, 1);   // never exceeds BUFCAP
                    bufv[p] = v;
                    bufm[p] = ((unsigned)m << 26) | (unsigned)col;
                }
            }
        }
        __syncthreads();

        // Compaction: wave w owns rows {2w, 2w+1}
        const int total = min(cnt, BUFCAP);
        for (int base = 0; base < total; base += 32) {
            const int i = base + lane;
            float v = 0.f; unsigned meta = 0u; bool have = false;
            if (i < total) {
                v = bufv[i]; meta = bufm[i];
                have = ((meta >> 27) == (unsigned)wave);   // (row>>1) == wave
            }
            unsigned long long bal = __ballot(have);
            while (bal) {
                const int src = __ffsll(bal) - 1;
                bal &= bal - 1;
                const float    cv = __shfl(v, src, 32);
                const unsigned cm = (unsigned)__shfl((int)meta, src, 32);
                const int rm = (int)(cm >> 26);
                if (cv > lmin[rm]) {                 // re-check vs fresh threshold
                    if (lane == 0) {
                        const int p = lminpos[rm];
                        lv[rm][p] = cv;
                        li[rm][p] = cm & 0x3FFFFFFu;
                    }
                    __builtin_amdgcn_wave_barrier();
                    // wave-parallel rescan of 128 entries for new min
                    float mv = lv[rm][lane]; int mp = lane;
#pragma unroll
                    for (int s = 1; s < 4; ++s) {
                        const float w2 = lv[rm][lane + 32 * s];
                        if (w2 < mv) { mv = w2; mp = lane + 32 * s; }
                    }
#pragma unroll
                    for (int off = 16; off > 0; off >>= 1) {
                        const float ov = __shfl_xor(mv, off, 32);
                        const int   op = __shfl_xor(mp, off, 32);
                        if (ov < mv) { mv = ov; mp = op; }
                    }
                    if (lane == 0) { lmin[rm] = mv; lminpos[rm] = mp; }
                    __builtin_amdgcn_wave_barrier();
                }
            }
        }
        __syncthreads();
        if (tid == 0) cnt = 0;
        __syncthreads();
    }

    // Emit per-(row, segment) 128-best lists
    for (int i = tid; i < 16 * LISTN; i += 256) {
        const int m = i >> 7, j = i & 127;
        const size_t o = ((size_t)(rowTile * 16 + m) * NSEG + seg) * LISTN + j;
        cand_val[o] = lv[m][j];
        cand_idx[o] = li[m][j];
    }
}

// ---------------------------------------------------------------------------
// Kernel 2: per row, merge NSEG*128 = 4096 candidates via in-LDS bitonic sort
// (descending; ties -> lower index, matching jax.lax.top_k), emit top-100.
// ---------------------------------------------------------------------------
__global__ __launch_bounds__(256, 1)
void topk_merge(const float* __restrict__ cand_val,
                const unsigned* __restrict__ cand_idx,
                float* __restrict__ out, int B, int TOPK)
{
    const int row = blockIdx.x;
    const int tid = threadIdx.x;
    const int M   = NSEG * LISTN;   // 4096, power of two

    __shared__ float    sv[NSEG * LISTN];
    __shared__ unsigned si[NSEG * LISTN];

    for (int i = tid; i < M; i += 256) {
        sv[i] = cand_val[(size_t)row * M + i];
        si[i] = cand_idx[(size_t)row * M + i];
    }
    __syncthreads();

    for (int k = 2; k <= M; k <<= 1) {
        for (int j = k >> 1; j > 0; j >>= 1) {
            for (int i = tid; i < M; i += 256) {
                const int ixj = i ^ j;
                if (ixj > i) {
                    const float    a  = sv[i], b = sv[ixj];
                    const unsigned ia = si[i], ib = si[ixj];
                    const bool a_better = (a > b) || (a == b && ia < ib);
                    const bool desc = ((i & k) == 0);
                    if (desc ? !a_better : a_better) {
                        sv[i] = b; sv[ixj] = a;
                        si[i] = ib; si[ixj] = ia;
                    }
                }
            }
            __syncthreads();
        }
    }

    if (tid < TOPK) {
        out[(size_t)row * TOPK + tid] = sv[tid];                           // scores
        out[(size_t)B * TOPK + (size_t)row * TOPK + tid] = (float)si[tid]; // indices
    }
}

// ---------------------------------------------------------------------------
extern "C" void kernel_launch(void* const* d_in, const int* in_sizes, int n_in,
                              void* d_out, int out_size, void* d_ws, size_t ws_size,
                              hipStream_t stream)
{
    (void)n_in; (void)out_size; (void)ws_size;

    const float* user   = (const float*)d_in[0];   // [B, 64] f32
    const float* corpus = (const float*)d_in[1];   // [N, 64] f32

    const int D = 64;
    const int B = in_sizes[0] / D;                 // 256
    const int N = in_sizes[1] / D;                 // 1,000,000
    const int TOPK = 100;
    const int segcols = (N + NSEG - 1) / NSEG;     // 31250

    float*    cand_val = (float*)d_ws;
    unsigned* cand_idx = (unsigned*)((char*)d_ws +
                         (size_t)B * NSEG * LISTN * sizeof(float));

    dim3 g1(B / 16, NSEG);   // row-tile fastest -> same-segment WGs co-resident (L2 reuse)
    hipLaunchKernelGGL(topk_gemm_partial, g1, dim3(256), 0, stream,
                       user, corpus, cand_val, cand_idx, N, segcols);
    hipLaunchKernelGGL(topk_merge, dim3(B), dim3(256), 0, stream,
                       cand_val, cand_idx, (float*)d_out, B, TOPK);
}